// ALS_RHMF_24524263260343
// MI455X (gfx1250) — compile-verified
//
#include <hip/hip_runtime.h>
#include <math.h>

typedef __attribute__((ext_vector_type(2))) float v2f;
typedef __attribute__((ext_vector_type(8))) float v8f;

#define DCOLS 512
#define NSEG 8
#define NU_F 5.0f
#define RIDGE_F 1e-6f
#define EPS_F 1e-6f
#define V8Z {0.0f,0.0f,0.0f,0.0f,0.0f,0.0f,0.0f,0.0f}

static __device__ __forceinline__ v8f wmma4(v2f a, v2f b, v8f c) {
  // D(16x16,f32) = A(16x4,f32) x B(4x16,f32) + C
  return __builtin_amdgcn_wmma_f32_16x16x4_f32(false, a, false, b, (short)0, c, false, false);
}

static __device__ __forceinline__ float fast_rcp(float x)  { return __builtin_amdgcn_rcpf(x); }
static __device__ __forceinline__ float fast_sqrt(float x) { return __builtin_amdgcn_sqrtf(x); }
static __device__ __forceinline__ float fast_log1p(float x) {
  return 0.69314718056f * __builtin_amdgcn_logf(1.0f + x);   // v_log_f32 is log2
}

// Cooperative 16x16 Gauss-Jordan on augmented [M|b] stored in LDS, stride 18.
// SPD + ridge -> no pivoting. All waves in the block call this uniformly.
static __device__ __forceinline__ void gj_solve16(float* aug, float* fcol, int l) {
  for (int j = 0; j < 16; ++j) {
    __syncthreads();
    float invp = fast_rcp(aug[j*18 + j]);
    if (l < 17) aug[j*18 + l] *= invp;      // scale pivot row (incl. rhs)
    if (l < 16) fcol[l] = aug[l*18 + j];    // capture column-j factors
    __syncthreads();
#pragma unroll
    for (int t = 0; t < 9; ++t) {
      int e = l + 32*t;                      // e < 288 enumerates r*18+c
      int r = e / 18;
      int c = e - r*18;
      if (r != j && c < 17) aug[e] -= fcol[r] * aug[j*18 + c];
    }
  }
  __syncthreads();
}

// ---------------- A-step: per-row weighted normal equations --------------
// block = 512 threads (16 waves) handles 16 rows.
__global__ __launch_bounds__(512) void a_step_kernel(
    const float* __restrict__ X, const float* __restrict__ W,
    const float* __restrict__ A, const float* __restrict__ G,
    float* __restrict__ Anew) {
  extern __shared__ float sm[];
  float* GT   = sm;                 // 16 x 516 : G transposed (GT[k*516+d]=G[d,k])
  float* SQXW = sm + 16*516;        // 16 rows x 512 x (sqw,xw) interleaved, stride 1028
  float* ABLK = SQXW + 16*1028;     // 16 x 18 : A tile
  float* GJ   = ABLK + 16*18;       // 16 waves x 304
  const int t  = threadIdx.x;
  const int wv = t >> 5;
  const int l  = t & 31;
  const int hf = l >> 4;
  const int lm = l & 15;
  const int row0 = blockIdx.x * 16;

  for (int e = t; e < DCOLS*16; e += 512) {
    int d = e >> 4, k = e & 15;
    GT[k*516 + d] = G[e];
  }
  for (int e = t; e < 256; e += 512)
    ABLK[(e>>4)*18 + (e&15)] = A[row0*16 + e];
  __syncthreads();

  // phase 1: resid tile P = A_tile(16x16) @ G^T, per wave 2 column-blocks of 16
  for (int cbi = 0; cbi < 2; ++cbi) {
    const int c0 = (2*wv + cbi) * 16;
    v8f p = V8Z;
#pragma unroll
    for (int kk0 = 0; kk0 < 16; kk0 += 4) {
      const int kk = kk0 + 2*hf;
      v2f av; av.x = ABLK[lm*18 + kk];        av.y = ABLK[lm*18 + kk + 1];
      v2f bv; bv.x = GT[kk*516 + c0 + lm];    bv.y = GT[(kk+1)*516 + c0 + lm];
      p = wmma4(av, bv, p);
    }
#pragma unroll
    for (int j = 0; j < 8; ++j) {
      const int i = j + 8*hf;
      const int d = c0 + lm;
      const int g = (row0 + i)*DCOLS + d;
      float x = X[g], w = W[g];
      float r = x - p[j];
      float weff = w * fast_rcp(1.0f + w*r*r*(1.0f/NU_F));
      float sq = fast_sqrt(weff);
      SQXW[i*1028 + 2*d]     = sq;
      SQXW[i*1028 + 2*d + 1] = sq * x;
    }
  }
  __syncthreads();

  // phase 2: wave wv owns row (row0+wv). M = Gw^T Gw, b = Gw^T xw via WMMA.
  const float* sx = SQXW + wv*1028;
  v8f cM = V8Z, cB = V8Z;
  for (int d0 = 0; d0 < DCOLS; d0 += 4) {
    const int d = d0 + 2*hf;
    float g0 = GT[lm*516 + d];
    float g1 = GT[lm*516 + d + 1];
    float s0 = sx[2*d], x0 = sx[2*d+1], s1 = sx[2*d+2], x1 = sx[2*d+3];
    v2f gw; gw.x = g0*s0; gw.y = g1*s1;
    v2f u;  u.x = x0;     u.y = x1;        // xw broadcast B-tile -> b replicated
    cM = wmma4(gw, gw, cM);
    cB = wmma4(gw, u,  cB);
  }

  float* aug  = GJ + wv*304;
  float* fcol = aug + 288;
#pragma unroll
  for (int j = 0; j < 8; ++j) {
    const int m = j + 8*hf;
    aug[m*18 + lm] = cM[j] + ((m == lm) ? RIDGE_F : 0.0f);
  }
  if (lm == 0) {
#pragma unroll
    for (int j = 0; j < 8; ++j) aug[(j + 8*hf)*18 + 16] = cB[j];
  }
  gj_solve16(aug, fcol, l);
  if (l < 16) Anew[(row0 + wv)*16 + l] = aug[l*18 + 16];
}

// ------------- G-step partials: per-column Gram over N, segmented ----------
__global__ __launch_bounds__(512) void g_partial_kernel(
    const float* __restrict__ X, const float* __restrict__ W,
    const float* __restrict__ A, const float* __restrict__ G,
    const float* __restrict__ Anew, float* __restrict__ part, int N) {
  __shared__ float GB[16*18];       // G block rows j0..j0+15
  __shared__ float ATN[16*260];     // A_new chunk transposed: ATN[k*260+i], i<256
  __shared__ float SQXWt[16*516];   // per column j: (sqw,xw) over 256 rows
  const int t = threadIdx.x, wv = t>>5, l = t&31, hf = l>>4, lm = l&15;
  const int j0 = blockIdx.x * 16;
  const int NC = N >> 4;                       // 16-row chunks total
  const int cps = (NC + NSEG - 1) / NSEG;
  const int cb = (int)blockIdx.y * cps;
  const int ce = min(NC, cb + cps);

  for (int e = t; e < 256; e += 512) GB[(e>>4)*18 + (e&15)] = G[j0*16 + e];
  __syncthreads();

  v8f cM = V8Z, cBv = V8Z;
  for (int g0 = cb; g0 < ce; g0 += 16) {       // group of up to 16 chunks (256 rows)
    const int nch = min(16, ce - g0);
    const int rows = nch * 16;
    const int R0 = g0 * 16;
    for (int e = t; e < rows*16; e += 512)
      ATN[(e&15)*260 + (e>>4)] = Anew[R0*16 + e];
    if (wv < nch) {                            // phase 1: chunk wv resid + weights
      const int r0 = R0 + wv*16;
      v8f p = V8Z;
#pragma unroll
      for (int kk0 = 0; kk0 < 16; kk0 += 4) {
        const int kk = kk0 + 2*hf;
        v2f av; av.x = A[(r0+lm)*16 + kk]; av.y = A[(r0+lm)*16 + kk + 1];
        v2f bv; bv.x = GB[lm*18 + kk];     bv.y = GB[lm*18 + kk + 1];
        p = wmma4(av, bv, p);
      }
#pragma unroll
      for (int j = 0; j < 8; ++j) {
        const int i = j + 8*hf;
        const int gi = (r0 + i)*DCOLS + j0 + lm;
        float x = X[gi], w = W[gi];
        float r = x - p[j];
        float weff = w * fast_rcp(1.0f + w*r*r*(1.0f/NU_F));
        float sq = fast_sqrt(weff);
        const int iloc = wv*16 + i;
        SQXWt[lm*516 + 2*iloc]     = sq;
        SQXWt[lm*516 + 2*iloc + 1] = sq * x;
      }
    }
    __syncthreads();
    // phase 2: wave wv = column j0+wv; contract over group rows
    const float* sx = SQXWt + wv*516;
    for (int i0 = 0; i0 < rows; i0 += 4) {
      const int i = i0 + 2*hf;
      float a0 = ATN[lm*260 + i];
      float a1 = ATN[lm*260 + i + 1];
      float s0 = sx[2*i], x0 = sx[2*i+1], s1 = sx[2*i+2], x1 = sx[2*i+3];
      v2f aw; aw.x = a0*s0; aw.y = a1*s1;
      v2f u;  u.x = x0;     u.y = x1;
      cM  = wmma4(aw, aw, cM);
      cBv = wmma4(aw, u,  cBv);
    }
    __syncthreads();
  }
  float* dst = part + ((size_t)(j0 + wv) * NSEG + blockIdx.y) * 272;
#pragma unroll
  for (int j = 0; j < 8; ++j) dst[(j + 8*hf)*16 + lm] = cM[j];
  if (lm == 0) {
#pragma unroll
    for (int j = 0; j < 8; ++j) dst[256 + j + 8*hf] = cBv[j];
  }
}

__global__ __launch_bounds__(512) void g_solve_kernel(
    const float* __restrict__ part, float* __restrict__ Gnew) {
  __shared__ float GJ[16*304];
  const int t = threadIdx.x, wv = t>>5, l = t&31;
  float* aug  = GJ + wv*304;
  float* fcol = aug + 288;
  const int j = blockIdx.x*16 + wv;
  const float* src = part + (size_t)j * NSEG * 272;
  for (int e = l; e < 272; e += 32) {
    float s = 0.0f;
    for (int sg = 0; sg < NSEG; ++sg) s += src[sg*272 + e];
    int r, c;
    if (e < 256) { r = e >> 4; c = e & 15; if (r == c) s += RIDGE_F; }
    else         { r = e - 256; c = 16; }
    aug[r*18 + c] = s;
  }
  gj_solve16(aug, fcol, l);
  if (l < 16) Gnew[j*16 + l] = aug[l*18 + 16];
}

// ---------------- C = Anew^T Anew partial Grams (WMMA) --------------------
__global__ __launch_bounds__(32) void c_gram_kernel(
    const float* __restrict__ Anew, float* __restrict__ Cpart, int N) {
  const int l = threadIdx.x, hf = l>>4, lm = l&15;
  const int nch = N >> 2;                      // 4-row chunks
  const int per = (nch + (int)gridDim.x - 1) / (int)gridDim.x;
  const int c0 = (int)blockIdx.x * per;
  const int c1 = min(nch, c0 + per);
  v8f c = V8Z;
  for (int ch = c0; ch < c1; ++ch) {
    const int i = ch*4 + 2*hf;
    v2f a; a.x = Anew[i*16 + lm]; a.y = Anew[(i+1)*16 + lm];
    c = wmma4(a, a, c);
  }
  float* dst = Cpart + (size_t)blockIdx.x * 256;
#pragma unroll
  for (int j = 0; j < 8; ++j) dst[(j + 8*hf)*16 + lm] = c[j];
}

// ------------- 16x16 eigh via cyclic Jacobi (one wave), R = C^{-1/2} ------
__global__ __launch_bounds__(32) void eig_kernel(
    const float* __restrict__ Cpart, int nparts,
    float* __restrict__ R, float* __restrict__ Rinv) {
  __shared__ float Cm[16*17];
  __shared__ float Vv[16*17];
  const int l = threadIdx.x;
  const int k = l & 15;
  for (int e = l; e < 256; e += 32) {
    float s = 0.0f;
    for (int b = 0; b < nparts; ++b) s += Cpart[(size_t)b*256 + e];
    int r = e >> 4, c = e & 15;
    Cm[r*17 + c] = s + ((r == c) ? EPS_F : 0.0f);
    Vv[r*17 + c] = (r == c) ? 1.0f : 0.0f;
  }
  __syncthreads();
  for (int sweep = 0; sweep < 12; ++sweep) {
    for (int p = 0; p < 15; ++p) {
      for (int q = p + 1; q < 16; ++q) {
        __syncthreads();
        float apq = Cm[p*17 + q];
        float app = Cm[p*17 + p];
        float aqq = Cm[q*17 + q];
        float cth = 1.0f, sth = 0.0f;
        if (fabsf(apq) > 1e-24f) {
          float tau = (aqq - app) / (2.0f * apq);
          float tt = ((tau >= 0.0f) ? 1.0f : -1.0f) /
                     (fabsf(tau) + sqrtf(1.0f + tau*tau));
          cth = 1.0f / sqrtf(1.0f + tt*tt);
          sth = tt * cth;
        }
        if (l < 16) {                       // rows p,q : B = J^T A
          float cpk = Cm[p*17 + k], cqk = Cm[q*17 + k];
          Cm[p*17 + k] = cth*cpk - sth*cqk;
          Cm[q*17 + k] = sth*cpk + cth*cqk;
        } else {                            // eigenvectors
          float vkp = Vv[k*17 + p], vkq = Vv[k*17 + q];
          Vv[k*17 + p] = cth*vkp - sth*vkq;
          Vv[k*17 + q] = sth*vkp + cth*vkq;
        }
        __syncthreads();
        if (l < 16) {                       // cols p,q : A' = B J
          float ckp = Cm[k*17 + p], ckq = Cm[k*17 + q];
          Cm[k*17 + p] = cth*ckp - sth*ckq;
          Cm[k*17 + q] = sth*ckp + cth*ckq;
        }
      }
    }
  }
  __syncthreads();
  for (int e = l; e < 256; e += 32) {
    int a = e >> 4, b = e & 15;
    float rs = 0.0f, ris = 0.0f;
    for (int kk = 0; kk < 16; ++kk) {
      float lam = fmaxf(Cm[kk*17 + kk], EPS_F);
      float sq = sqrtf(lam);
      float pr = Vv[a*17 + kk] * Vv[b*17 + kk];
      rs  += pr / sq;
      ris += pr * sq;
    }
    R[e] = rs;
    Rinv[e] = ris;
  }
}

__global__ void g2_kernel(const float* __restrict__ Gnew,
                          const float* __restrict__ Rinv,
                          float* __restrict__ G2) {
  const int idx = blockIdx.x * blockDim.x + threadIdx.x;
  if (idx < DCOLS*16) {
    const int r = idx >> 4, c = idx & 15;
    float s = 0.0f;
#pragma unroll
    for (int kk = 0; kk < 16; ++kk) s += Gnew[r*16 + kk] * Rinv[kk*16 + c];
    G2[idx] = s;
  }
}

// ------------- A2 = Anew @ R (WMMA) + per-block robust-loss partial -------
__global__ __launch_bounds__(512) void apply_loss_kernel(
    const float* __restrict__ X, const float* __restrict__ W,
    const float* __restrict__ Anew, const float* __restrict__ Gnew,
    const float* __restrict__ R, float* __restrict__ A2,
    float* __restrict__ losspart) {
  __shared__ float red[512];
  const int t = threadIdx.x, wv = t>>5, l = t&31, hf = l>>4, lm = l&15;
  const int row0 = blockIdx.x * 16;
  if (wv == 0) {
    v8f p = V8Z;
#pragma unroll
    for (int kk0 = 0; kk0 < 16; kk0 += 4) {
      const int kk = kk0 + 2*hf;
      v2f av; av.x = Anew[(row0+lm)*16 + kk]; av.y = Anew[(row0+lm)*16 + kk + 1];
      v2f bv; bv.x = R[kk*16 + lm];           bv.y = R[(kk+1)*16 + lm];
      p = wmma4(av, bv, p);
    }
#pragma unroll
    for (int j = 0; j < 8; ++j) A2[(row0 + j + 8*hf)*16 + lm] = p[j];
  }
  float acc = 0.0f;
  for (int cbi = 0; cbi < 2; ++cbi) {
    const int c0 = (2*wv + cbi) * 16;
    v8f p = V8Z;
#pragma unroll
    for (int kk0 = 0; kk0 < 16; kk0 += 4) {
      const int kk = kk0 + 2*hf;
      v2f av; av.x = Anew[(row0+lm)*16 + kk]; av.y = Anew[(row0+lm)*16 + kk + 1];
      v2f bv; bv.x = Gnew[(c0+lm)*16 + kk];   bv.y = Gnew[(c0+lm)*16 + kk + 1];
      p = wmma4(av, bv, p);
    }
#pragma unroll
    for (int j = 0; j < 8; ++j) {
      const int gi = (row0 + j + 8*hf)*DCOLS + c0 + lm;
      float x = X[gi], w = W[gi];
      float r = x - p[j];
      acc += fast_log1p(w*r*r*(1.0f/NU_F));
    }
  }
  red[t] = acc;
  __syncthreads();
  for (int s = 256; s > 0; s >>= 1) {
    if (t < s) red[t] += red[t + s];
    __syncthreads();
  }
  if (t == 0) losspart[blockIdx.x] = red[0];
}

__global__ __launch_bounds__(512) void loss_reduce_kernel(
    const float* __restrict__ lp, float* __restrict__ out, int n) {
  __shared__ float red[512];
  const int t = threadIdx.x;
  float s = 0.0f;
  for (int i = t; i < n; i += 512) s += lp[i];
  red[t] = s;
  __syncthreads();
  for (int k = 256; k > 0; k >>= 1) {
    if (t < k) red[t] += red[t + k];
    __syncthreads();
  }
  if (t == 0) out[0] = 0.5f * NU_F * red[0];
}

extern "C" void kernel_launch(void* const* d_in, const int* in_sizes, int n_in,
                              void* d_out, int out_size, void* d_ws, size_t ws_size,
                              hipStream_t stream) {
  const float* X = (const float*)d_in[0];
  const float* W = (const float*)d_in[1];
  const float* A = (const float*)d_in[2];
  const float* G = (const float*)d_in[3];
  const int N = in_sizes[2] / 16;          // 100000
  const int nblk = N / 16;                 // 6250 (N divisible by 16)
  (void)n_in; (void)ws_size; (void)out_size;

  float* ws = (float*)d_ws;
  float* Anew     = ws;                              // N*16
  float* Gnew     = Anew + (size_t)N*16;             // 8192
  float* Rm       = Gnew + DCOLS*16;                 // 256
  float* Rinv     = Rm + 256;                        // 256
  float* Cpart    = Rinv + 256;                      // 128*256
  float* losspart = Cpart + 128*256;                 // nblk
  float* part     = losspart + nblk;                 // DCOLS*NSEG*272

  float* out = (float*)d_out;
  float* A2    = out;
  float* G2    = out + (size_t)N*16;
  float* lossp = G2 + DCOLS*16;

  const int aLds = (16*516 + 16*1028 + 16*18 + 16*304) * 4;  // ~119 KB dynamic LDS
  hipFuncSetAttribute((const void*)a_step_kernel,
                      hipFuncAttributeMaxDynamicSharedMemorySize, aLds);

  a_step_kernel<<<nblk, 512, aLds, stream>>>(X, W, A, G, Anew);
  g_partial_kernel<<<dim3(DCOLS/16, NSEG), 512, 0, stream>>>(X, W, A, G, Anew, part, N);
  g_solve_kernel<<<DCOLS/16, 512, 0, stream>>>(part, Gnew);
  c_gram_kernel<<<128, 32, 0, stream>>>(Anew, Cpart, N);
  eig_kernel<<<1, 32, 0, stream>>>(Cpart, 128, Rm, Rinv);
  g2_kernel<<<(DCOLS*16 + 255)/256, 256, 0, stream>>>(Gnew, Rinv, G2);
  apply_loss_kernel<<<nblk, 512, 0, stream>>>(X, W, Anew, Gnew, Rm, A2, losspart);
  loss_reduce_kernel<<<1, 512, 0, stream>>>(losspart, lossp, nblk);
}